// MoEGate_63969242906686
// MI455X (gfx1250) — compile-verified
//
#include <hip/hip_runtime.h>
#include <hip/hip_bf16.h>
#include <math.h>

#define NUM_EXPERTS 64
#define TOPK 6
#define HIDDEN 128
#define ALPHA 1e-3f

typedef float v2f __attribute__((ext_vector_type(2)));
typedef float v4f __attribute__((ext_vector_type(4)));
typedef float v8f __attribute__((ext_vector_type(8)));

#define WAVES_PER_WG 4
#define TOKENS_PER_WAVE 32
#define TOKENS_PER_WG (WAVES_PER_WG * TOKENS_PER_WAVE) // 128
#define XPAD 132   // LDS row stride in floats: 132 mod 64 = 4 -> conflict-free b64 frag reads
#define LROW 68    // logits scratch row stride (within reused X slice)

// ---------------------------------------------------------------------------
// Logits via V_WMMA_F32_16X16X4_F32 with LDS-staged, coalesced data movement.
// ---------------------------------------------------------------------------
__global__ __launch_bounds__(TOKENS_PER_WG)
void moe_gate_kernel(const float* __restrict__ X, const float* __restrict__ W,
                     int* __restrict__ idx_out, float* __restrict__ w_out,
                     float* __restrict__ pi_acc, float* __restrict__ cnt_acc,
                     int N)
{
    __shared__ float lds_x[TOKENS_PER_WG * XPAD];   // 128 tokens x 128 (padded)  ~67.6 KB
    __shared__ float lds_w[NUM_EXPERTS * XPAD];     // 64 experts x 128 (padded)  ~33.8 KB
    __shared__ float lds_pi[NUM_EXPERTS];
    __shared__ float lds_cnt[NUM_EXPERTS];

    const int tid  = threadIdx.x;
    const int wave = tid >> 5;
    const int lane = tid & 31;

    if (tid < NUM_EXPERTS) { lds_pi[tid] = 0.0f; lds_cnt[tid] = 0.0f; }

    // ---- Coalesced staging: consecutive threads load consecutive 16B ----
    const int base_tok = blockIdx.x * TOKENS_PER_WG;
    #pragma unroll 4
    for (int i = 0; i < 32; ++i) {              // 128x128 f32 = 4096 v4f
        const int e = i * TOKENS_PER_WG + tid;
        const int r = e >> 5;                   // token row in block (32 v4f per row)
        const int c = e & 31;
        int gr = base_tok + r; if (gr > N - 1) gr = N - 1;
        v4f v = *(const v4f*)(X + (size_t)gr * HIDDEN + c * 4);
        *(v4f*)(lds_x + r * XPAD + c * 4) = v;
    }
    #pragma unroll 4
    for (int i = 0; i < 16; ++i) {              // 64x128 f32 = 2048 v4f
        const int e = i * TOKENS_PER_WG + tid;
        const int r = e >> 5;
        const int c = e & 31;
        v4f v = *(const v4f*)(W + (size_t)r * HIDDEN + c * 4);
        *(v4f*)(lds_w + r * XPAD + c * 4) = v;
    }
    __syncthreads();

    // ---- WMMA GEMM: D(16 experts x 16 tokens) = W-tile(16x4) * xT(4x16) ----
    const int tok_base = base_tok + wave * TOKENS_PER_WAVE;
    const int lane_lo  = lane & 15;
    const int half     = lane >> 4;        // 0: K=0,1   1: K=2,3 of each 4-chunk
    const int fo       = half * 2;

    float* slice = lds_x + (wave * TOKENS_PER_WAVE) * XPAD;   // this wave's 32 tokens
    const float* xb0 = slice + lane_lo * XPAD + fo;
    const float* xb1 = slice + (16 + lane_lo) * XPAD + fo;
    const float* wb  = lds_w + lane_lo * XPAD + fo;

    v8f acc[2][4] = {};

    #pragma unroll 4
    for (int c = 0; c < 32; ++c) {
        const int ko = c * 4;
        v2f b0 = *(const v2f*)(xb0 + ko);
        v2f b1 = *(const v2f*)(xb1 + ko);
        #pragma unroll
        for (int t = 0; t < 4; ++t) {
            v2f a = *(const v2f*)(wb + t * 16 * XPAD + ko);
            acc[0][t] = __builtin_amdgcn_wmma_f32_16x16x4_f32(
                false, a, false, b0, (short)0, acc[0][t], false, false);
            acc[1][t] = __builtin_amdgcn_wmma_f32_16x16x4_f32(
                false, a, false, b1, (short)0, acc[1][t], false, false);
        }
    }

    // ---- Transpose accumulators to token-major rows (reuse own X slice) ----
    // Lane n (n<16) holds experts 16t+r for token n; lane n+16 holds 16t+8+r.
    #pragma unroll
    for (int g = 0; g < 2; ++g) {
        float* rp = slice + (g * 16 + lane_lo) * LROW + 8 * half;
        #pragma unroll
        for (int t = 0; t < 4; ++t) {
            v4f lo = { acc[g][t][0], acc[g][t][1], acc[g][t][2], acc[g][t][3] };
            v4f hi = { acc[g][t][4], acc[g][t][5], acc[g][t][6], acc[g][t][7] };
            *(v4f*)(rp + 16 * t)     = lo;
            *(v4f*)(rp + 16 * t + 4) = hi;
        }
    }

    // ---- One lane per token: softmax + top-6 ----
    const int my_tok = tok_base + lane;
    float s[NUM_EXPERTS];
    {
        const float* rrow = slice + lane * LROW;
        #pragma unroll
        for (int i = 0; i < 16; ++i) {
            v4f v = *(const v4f*)(rrow + 4 * i);
            s[4*i+0] = v[0]; s[4*i+1] = v[1]; s[4*i+2] = v[2]; s[4*i+3] = v[3];
        }
    }
    float m = s[0];
    #pragma unroll
    for (int i = 1; i < NUM_EXPERTS; ++i) m = fmaxf(m, s[i]);
    float Z = 0.0f;
    #pragma unroll
    for (int i = 0; i < NUM_EXPERTS; ++i) { s[i] = __expf(s[i] - m); Z += s[i]; }
    const float invZ = 1.0f / Z;

    // Write normalized scores back for per-expert column sums.
    {
        float* rrow = slice + lane * LROW;
        #pragma unroll
        for (int i = 0; i < 16; ++i) {
            v4f v = { s[4*i+0]*invZ, s[4*i+1]*invZ, s[4*i+2]*invZ, s[4*i+3]*invZ };
            *(v4f*)(rrow + 4 * i) = v;
        }
    }

    // Per-expert partials: lane sums 2 adjacent score columns of its wave's tokens.
    {
        float cs0 = 0.0f, cs1 = 0.0f;
        #pragma unroll 8
        for (int r = 0; r < TOKENS_PER_WAVE; ++r) {
            v2f v = *(const v2f*)(slice + r * LROW + 2 * lane);
            cs0 += v[0]; cs1 += v[1];
        }
        atomicAdd(&lds_pi[2 * lane + 0], cs0);
        atomicAdd(&lds_pi[2 * lane + 1], cs1);
    }

    // Top-6 by masked argmax (monotonic in score); ties -> lowest index.
    if (my_tok < N) {
        float bval[TOPK]; int bidx[TOPK];
        float wsum = 0.0f;
        #pragma unroll
        for (int j = 0; j < TOPK; ++j) {
            float best = -1.0f; int bi = 0;   // scores are > 0
            #pragma unroll
            for (int i = 0; i < NUM_EXPERTS; ++i) {
                bool gt = s[i] > best;
                best = gt ? s[i] : best;
                bi   = gt ? i    : bi;
            }
            bval[j] = best; bidx[j] = bi; wsum += best;
            #pragma unroll
            for (int i = 0; i < NUM_EXPERTS; ++i) s[i] = (i == bi) ? -1.0f : s[i];
        }
        const float invw = 1.0f / (wsum + 1e-20f);
        const size_t ob = (size_t)my_tok * TOPK;
        #pragma unroll
        for (int j = 0; j < TOPK; ++j) {
            idx_out[ob + j] = bidx[j];
            w_out[ob + j]   = bval[j] * invw;
            atomicAdd(&lds_cnt[bidx[j]], 1.0f);
        }
    }

    __syncthreads();
    if (tid < NUM_EXPERTS) {
        atomicAdd(&pi_acc[tid],  lds_pi[tid]);
        atomicAdd(&cnt_acc[tid], lds_cnt[tid]);
    }
}

// ---------------------------------------------------------------------------
// aux = ALPHA * 64/(6*N*N) * sum_e pi_sum[e] * counts[e]
// ---------------------------------------------------------------------------
__global__ __launch_bounds__(NUM_EXPERTS)
void moe_aux_kernel(const float* __restrict__ pi_acc,
                    const float* __restrict__ cnt_acc,
                    float* __restrict__ aux_out, float scale)
{
    __shared__ float red[NUM_EXPERTS];
    const int t = threadIdx.x;
    red[t] = pi_acc[t] * cnt_acc[t];
    __syncthreads();
    for (int sft = NUM_EXPERTS / 2; sft > 0; sft >>= 1) {
        if (t < sft) red[t] += red[t + sft];
        __syncthreads();
    }
    if (t == 0) *aux_out = red[0] * scale;
}

// ---------------------------------------------------------------------------
extern "C" void kernel_launch(void* const* d_in, const int* in_sizes, int n_in,
                              void* d_out, int out_size, void* d_ws, size_t ws_size,
                              hipStream_t stream)
{
    const float* X = (const float*)d_in[0];   // (64,4096,128) f32
    const float* W = (const float*)d_in[1];   // (64,128) f32
    const int N = in_sizes[0] / HIDDEN;       // 262144 tokens

    int*   idx_out = (int*)d_out;                              // N*6 int32
    float* w_out   = (float*)d_out + (size_t)N * TOPK;         // N*6 f32
    float* aux_out = (float*)d_out + (size_t)2 * N * TOPK;     // 1 f32

    float* pi_acc  = (float*)d_ws;
    float* cnt_acc = pi_acc + NUM_EXPERTS;
    hipMemsetAsync(d_ws, 0, 2 * NUM_EXPERTS * sizeof(float), stream);

    const int blocks = (N + TOKENS_PER_WG - 1) / TOKENS_PER_WG;
    moe_gate_kernel<<<blocks, TOKENS_PER_WG, 0, stream>>>(
        X, W, idx_out, w_out, pi_acc, cnt_acc, N);

    const double scale =
        (double)ALPHA * (double)NUM_EXPERTS /
        ((double)N * (double)TOPK * (double)N);
    moe_aux_kernel<<<1, NUM_EXPERTS, 0, stream>>>(pi_acc, cnt_acc, aux_out,
                                                  (float)scale);
}